// FusionMethodA_46703474376898
// MI455X (gfx1250) — compile-verified
//
#include <hip/hip_runtime.h>
#include <hip/hip_bf16.h>
#include <cstdint>

#define NN 50000
#define EE 800000
#define DD 128
#define NLAYER 2
#define BN_EPS 1e-5f

typedef __attribute__((ext_vector_type(2))) float v2f;
typedef __attribute__((ext_vector_type(8))) float v8f;

// ---------------- degree / symmetric norm ----------------
__global__ void k_init_deg(float* deg) {
    int i = blockIdx.x * blockDim.x + threadIdx.x;
    if (i < NN) deg[i] = 1.0f;               // self-loop weight 1
}

__global__ void k_accum_deg(const long long* __restrict__ dst,
                            const float* __restrict__ ew,
                            float* __restrict__ deg) {
    int e = blockIdx.x * blockDim.x + threadIdx.x;
    if (e < EE) unsafeAtomicAdd(&deg[dst[e]], ew[e]);
}

__global__ void k_dinv(float* deg) {
    int i = blockIdx.x * blockDim.x + threadIdx.x;
    if (i < NN) {
        float d = deg[i];
        deg[i] = (d > 0.0f) ? rsqrtf(d) : 0.0f;
    }
}

// ---------------- fp32 WMMA GEMM: C[N x 128] = A[N x 128] @ W[128 x 128] ----------------
// block = 256 threads (8 waves). Block covers 16 rows of A; wave w covers columns [16w,16w+16).
__global__ __launch_bounds__(256)
void k_gemm_wmma(const float* __restrict__ A, const float* __restrict__ W,
                 float* __restrict__ C) {
    __shared__ float sA[16 * 132];           // pitch 132 to avoid 64-bank conflicts
    const int m0 = blockIdx.x * 16;

    for (int j = threadIdx.x; j < 16 * 128; j += 256) {
        int r = j >> 7, c = j & 127;
        sA[r * 132 + c] = A[(size_t)(m0 + r) * DD + c];
    }
    __syncthreads();

    const int wave = threadIdx.x >> 5;
    const int lane = threadIdx.x & 31;
    const int hi   = lane >> 4;              // 0: K pair (k0,k0+1); 1: (k0+2,k0+3)
    const int lo   = lane & 15;
    const int n    = wave * 16 + lo;

    v8f acc = {};
#pragma unroll
    for (int k0 = 0; k0 < DD; k0 += 4) {
        const int kb = k0 + 2 * hi;
        v2f a = *(const v2f*)&sA[lo * 132 + kb];
        v2f b;
        b.x = W[(size_t)kb * DD + n];
        b.y = W[(size_t)(kb + 1) * DD + n];
        acc = __builtin_amdgcn_wmma_f32_16x16x4_f32(false, a, false, b,
                                                    (short)0, acc, false, false);
    }
#pragma unroll
    for (int i = 0; i < 8; ++i)
        C[(size_t)(m0 + i + 8 * hi) * DD + n] = acc[i];
}

// ---------------- aggregation ----------------
// out[i][d] = hw[i][d]*dinv[i]^2 (self loop) + bias[d]
__global__ void k_agg_init(const float* __restrict__ hw, const float* __restrict__ dinv,
                           const float* __restrict__ bias, float* __restrict__ out) {
    int idx = blockIdx.x * blockDim.x + threadIdx.x;
    if (idx < NN * DD) {
        int i = idx >> 7, d = idx & 127;
        float di = dinv[i];
        out[idx] = hw[idx] * di * di + bias[d];
    }
}

// one wave32 per edge: gather hw[src] (float4/lane), scale, atomic scatter into out[dst]
__global__ __launch_bounds__(256)
void k_agg_edges(const float* __restrict__ hw, const long long* __restrict__ src,
                 const long long* __restrict__ dst, const float* __restrict__ ew,
                 const float* __restrict__ dinv, float* __restrict__ out) {
    const int e    = (blockIdx.x * blockDim.x + threadIdx.x) >> 5;
    const int lane = threadIdx.x & 31;
    if (e >= EE) return;
    const long long s = src[e];
    const long long d = dst[e];
    const float nrm = dinv[s] * ew[e] * dinv[d];
    const float4 v = *(const float4*)&hw[(size_t)s * DD + lane * 4];
    float* o = &out[(size_t)d * DD + lane * 4];
    unsafeAtomicAdd(o + 0, v.x * nrm);
    unsafeAtomicAdd(o + 1, v.y * nrm);
    unsafeAtomicAdd(o + 2, v.z * nrm);
    unsafeAtomicAdd(o + 3, v.w * nrm);
}

// ---------------- batchnorm (train stats) + ReLU ----------------
__global__ void k_zero_stats(float* stats) {
    if (threadIdx.x < 2 * DD) stats[threadIdx.x] = 0.0f;
}

__global__ __launch_bounds__(128)
void k_col_stats(const float* __restrict__ h, float* __restrict__ stats) {
    const int d = threadIdx.x;               // one column per thread, coalesced rows
    float s = 0.0f, q = 0.0f;
    for (int r = blockIdx.x; r < NN; r += gridDim.x) {
        float v = h[(size_t)r * DD + d];
        s += v;
        q += v * v;
    }
    unsafeAtomicAdd(&stats[d], s);
    unsafeAtomicAdd(&stats[DD + d], q);
}

__global__ void k_bn_relu(float* __restrict__ h, const float* __restrict__ stats,
                          const float* __restrict__ gamma, const float* __restrict__ beta) {
    int idx = blockIdx.x * blockDim.x + threadIdx.x;
    if (idx < NN * DD) {
        int d = idx & 127;
        float mean = stats[d] * (1.0f / NN);
        float var  = stats[DD + d] * (1.0f / NN) - mean * mean;
        float v = (h[idx] - mean) * rsqrtf(var + BN_EPS);
        v = v * gamma[d] + beta[d];
        h[idx] = fmaxf(v, 0.0f);
    }
}

// ---------------- gated fusion: sigmoid([h_sc|h_fc] @ gW + gb), blend ----------------
__global__ __launch_bounds__(256)
void k_gate_wmma(const float* __restrict__ hsc, const float* __restrict__ hfc,
                 const float* __restrict__ gW, const float* __restrict__ gb,
                 float* __restrict__ out) {
    __shared__ float sS[16 * 132];
    __shared__ float sF[16 * 132];
    const int m0 = blockIdx.x * 16;

    for (int j = threadIdx.x; j < 16 * 128; j += 256) {
        int r = j >> 7, c = j & 127;
        sS[r * 132 + c] = hsc[(size_t)(m0 + r) * DD + c];
        sF[r * 132 + c] = hfc[(size_t)(m0 + r) * DD + c];
    }
    __syncthreads();

    const int wave = threadIdx.x >> 5;
    const int lane = threadIdx.x & 31;
    const int hi   = lane >> 4;
    const int lo   = lane & 15;
    const int n    = wave * 16 + lo;

    v8f acc = {};
#pragma unroll
    for (int k0 = 0; k0 < DD; k0 += 4) {      // K = 0..127 : h_sc part of gate_W
        const int kb = k0 + 2 * hi;
        v2f a = *(const v2f*)&sS[lo * 132 + kb];
        v2f b;
        b.x = gW[(size_t)kb * DD + n];
        b.y = gW[(size_t)(kb + 1) * DD + n];
        acc = __builtin_amdgcn_wmma_f32_16x16x4_f32(false, a, false, b,
                                                    (short)0, acc, false, false);
    }
#pragma unroll
    for (int k0 = 0; k0 < DD; k0 += 4) {      // K = 128..255 : h_fc part of gate_W
        const int kb = k0 + 2 * hi;
        v2f a = *(const v2f*)&sF[lo * 132 + kb];
        v2f b;
        b.x = gW[(size_t)(DD + kb) * DD + n];
        b.y = gW[(size_t)(DD + kb + 1) * DD + n];
        acc = __builtin_amdgcn_wmma_f32_16x16x4_f32(false, a, false, b,
                                                    (short)0, acc, false, false);
    }

    const float bias = gb[n];
#pragma unroll
    for (int i = 0; i < 8; ++i) {
        const int r = i + 8 * hi;            // row within the 16-row tile
        float g  = 1.0f / (1.0f + __expf(-(acc[i] + bias)));
        float hs = sS[r * 132 + n];
        float hf = sF[r * 132 + n];
        out[(size_t)(m0 + r) * DD + n] = g * hs + (1.0f - g) * hf;
    }
}

// ---------------- host orchestration ----------------
static void run_branch(const float* x, const long long* ei, const float* ew,
                       const float* W, const float* b, const float* g, const float* be,
                       float* bufA, float* hout, float* dinv, float* stats,
                       hipStream_t s) {
    k_init_deg <<<(NN + 255) / 256, 256, 0, s>>>(dinv);
    k_accum_deg<<<(EE + 255) / 256, 256, 0, s>>>(ei + EE, ew, dinv);
    k_dinv     <<<(NN + 255) / 256, 256, 0, s>>>(dinv);

    const float* hin = x;
    for (int l = 0; l < NLAYER; ++l) {
        k_gemm_wmma<<<NN / 16, 256, 0, s>>>(hin, W + (size_t)l * DD * DD, bufA);
        k_agg_init <<<(NN * DD) / 256, 256, 0, s>>>(bufA, dinv, b + l * DD, hout);
        k_agg_edges<<<(EE * 32) / 256, 256, 0, s>>>(bufA, ei, ei + EE, ew, dinv, hout);
        k_zero_stats<<<1, 256, 0, s>>>(stats);
        k_col_stats<<<1024, 128, 0, s>>>(hout, stats);
        k_bn_relu  <<<(NN * DD) / 256, 256, 0, s>>>(hout, stats, g + l * DD, be + l * DD);
        hin = hout;
    }
}

extern "C" void kernel_launch(void* const* d_in, const int* in_sizes, int n_in,
                              void* d_out, int out_size, void* d_ws, size_t ws_size,
                              hipStream_t stream) {
    (void)in_sizes; (void)n_in; (void)out_size; (void)ws_size;

    const float*     x     = (const float*)    d_in[0];
    const long long* ei_sc = (const long long*)d_in[1];
    const float*     ew_sc = (const float*)    d_in[2];
    const long long* ei_fc = (const long long*)d_in[3];
    const float*     ew_fc = (const float*)    d_in[4];
    const float*     W_sc  = (const float*)    d_in[5];
    const float*     b_sc  = (const float*)    d_in[6];
    const float*     g_sc  = (const float*)    d_in[7];
    const float*     be_sc = (const float*)    d_in[8];
    const float*     W_fc  = (const float*)    d_in[9];
    const float*     b_fc  = (const float*)    d_in[10];
    const float*     g_fc  = (const float*)    d_in[11];
    const float*     be_fc = (const float*)    d_in[12];
    const float*     gateW = (const float*)    d_in[13];
    const float*     gateb = (const float*)    d_in[14];

    float* ws = (float*)d_ws;
    const size_t ND = (size_t)NN * DD;
    float* bufA  = ws;                 // GEMM output scratch (shared by both branches)
    float* bufSC = ws + ND;            // sc branch hidden / final
    float* bufFC = ws + 2 * ND;        // fc branch hidden / final
    float* dinv  = ws + 3 * ND;        // N floats (deg -> dinv in place)
    float* stats = dinv + NN;          // 2*D floats (col sum, col sumsq)

    run_branch(x, ei_sc, ew_sc, W_sc, b_sc, g_sc, be_sc, bufA, bufSC, dinv, stats, stream);
    run_branch(x, ei_fc, ew_fc, W_fc, b_fc, g_fc, be_fc, bufA, bufFC, dinv, stats, stream);

    k_gate_wmma<<<NN / 16, 256, 0, stream>>>(bufSC, bufFC, gateW, gateb, (float*)d_out);
}